// META_WTN_SFTOutputs_17686675325289
// MI455X (gfx1250) — compile-verified
//
#include <hip/hip_runtime.h>
#include <hip/hip_bf16.h>
#include <stdint.h>

// FCOS target assignment for MI455X (gfx1250, wave32).
// Latency/output-bandwidth bound: ~0.4 G VALU-ops, ~7.8 MB of output writes.
// CDNA5 paths used: async global->LDS staging of the per-image box table
// (global_load_async_to_lds_b32 + s_wait_asynccnt), global_prefetch_b8 (WGP
// scope) for the streamed per-location loads, wave32-sized blocks,
// B64/B128 loads, B128 stores, 2x ds_load_b128 per inner iteration.

#define NBOX   50
#define BLOCK  256
#define INF_A  100000000.0f
#define NUM_CLASSES 80

#if __has_builtin(__builtin_amdgcn_global_load_async_to_lds_b32)
#define HAVE_ASYNC_LDS 1
#else
#define HAVE_ASYNC_LDS 0
#endif

typedef __attribute__((address_space(1))) int g_int;
typedef __attribute__((address_space(3))) int l_int;

__global__ __launch_bounds__(BLOCK) void fcos_assign_kernel(
    const float* __restrict__ locations,    // (L,2)
    const float* __restrict__ stride_loc,   // (L,)
    const float* __restrict__ size_ranges,  // (L,2)
    const float* __restrict__ gt_boxes,     // (B,NBOX,4)
    const int*   __restrict__ gt_classes,   // (B,NBOX)
    int*   __restrict__ out_labels,         // (B,L)
    float* __restrict__ out_reg,            // (B,L,4)
    float* __restrict__ out_ctr,            // (B,L)
    int L)
{
  __shared__ float4 s_box[NBOX];      // x0 y0 x1 y1
  __shared__ float4 s_der[NBOX];      // cx cy area class-bits
  __shared__ int    s_clsraw[NBOX];

  const int t = threadIdx.x;
  const int b = blockIdx.y;
  const int i = blockIdx.x * BLOCK + t;

  // Prefetch this thread's streamed per-location data into the WGP cache
  // (locality 3 -> WGP-scope global_prefetch_b8, fills all levels) while the
  // block stages the box table into LDS.
  if (i < L) {
    __builtin_prefetch(locations + 2 * (size_t)i, 0, 3);
    __builtin_prefetch(size_ranges + 2 * (size_t)i, 0, 3);
    __builtin_prefetch(stride_loc + i, 0, 3);
  }

  // Stage 50 boxes (200 dwords) + 50 classes (50 dwords) for image b into
  // LDS: one dword per lane, 250 active lanes, via the async datamover path.
  {
    const void* gsrc = nullptr;
    void* ldst = nullptr;
    if (t < NBOX * 4) {
      gsrc = gt_boxes + (size_t)b * (NBOX * 4) + t;
      ldst = (float*)s_box + t;
    } else if (t < NBOX * 5) {
      gsrc = gt_classes + (size_t)b * NBOX + (t - NBOX * 4);
      ldst = &s_clsraw[t - NBOX * 4];
    }
    if (gsrc) {
#if HAVE_ASYNC_LDS
      __builtin_amdgcn_global_load_async_to_lds_b32(
          (g_int*)(uintptr_t)gsrc,
          (l_int*)ldst,
          0, 0);
#else
      *(int*)ldst = *(const int*)gsrc;   // fallback: vmem load + ds_store
#endif
    }
  }
#if HAVE_ASYNC_LDS
#if __has_builtin(__builtin_amdgcn_s_wait_asynccnt)
  __builtin_amdgcn_s_wait_asynccnt(0);
#else
  asm volatile("s_wait_asynccnt 0x0" ::: "memory");
#endif
#endif
  __syncthreads();

  // Per-box derived values (center, area, class) packed into one float4.
  if (t < NBOX) {
    const float4 bx = s_box[t];
    float4 d;
    d.x = (bx.x + bx.z) * 0.5f;                 // cx
    d.y = (bx.y + bx.w) * 0.5f;                 // cy
    d.z = (bx.z - bx.x) * (bx.w - bx.y);        // area
    d.w = __int_as_float(s_clsraw[t]);          // class bits
    s_der[t] = d;
  }
  __syncthreads();

  if (i >= L) return;

  const float2 xy  = ((const float2*)locations)[i];    // global_load_b64
  const float2 rng = ((const float2*)size_ranges)[i];  // global_load_b64
  const float  sv  = stride_loc[i];
  const float  rad = sv * 1.5f;

  float best = INF_A;
  int   bj   = 0;
#pragma unroll 2
  for (int j = 0; j < NBOX; ++j) {
    const float4 bx = s_box[j];                 // ds_load_b128 (broadcast)
    const float4 d  = s_der[j];                 // ds_load_b128 (broadcast)
    const float l  = xy.x - bx.x;
    const float tt = xy.y - bx.y;
    const float r  = bx.z - xy.x;
    const float bb = bx.w - xy.y;
    const float gx0 = fmaxf(d.x - rad, bx.x);
    const float gy0 = fmaxf(d.y - rad, bx.y);
    const float gx1 = fminf(d.x + rad, bx.z);
    const float gy1 = fminf(d.y + rad, bx.w);
    const float cbmin = fminf(fminf(xy.x - gx0, xy.y - gy0),
                              fminf(gx1 - xy.x, gy1 - xy.y));
    const float maxreg = fminf(fmaxf(l, r), fmaxf(tt, bb));
    const bool ok = (cbmin > 0.0f) && (maxreg >= rng.x) && (maxreg <= rng.y);
    const float a = ok ? d.z : INF_A;
    if (a < best) { best = a; bj = j; }   // strict '<' == argmin first-index
  }

  // Gather best box; strides are powers of two so 1/sv*x == x/sv bit-exactly.
  const float4 bx = s_box[bj];
  const float inv = 1.0f / sv;
  const float l  = (xy.x - bx.x) * inv;
  const float tt = (xy.y - bx.y) * inv;
  const float r  = (bx.z - xy.x) * inv;
  const float bb = (bx.w - xy.y) * inv;

  const int label = (best >= INF_A) ? NUM_CLASSES
                                    : __float_as_int(s_der[bj].w);

  const float le = l + 1e-5f, re = r + 1e-5f;
  const float te = tt + 1e-5f, be = bb + 1e-5f;
  float c = (fminf(le, re) / fmaxf(le, re)) * (fminf(te, be) / fmaxf(te, be));
  c = sqrtf(fmaxf(c, 0.0f));

  const size_t o = (size_t)b * (size_t)L + (size_t)i;
  out_labels[o] = label;
  reinterpret_cast<float4*>(out_reg)[o] = make_float4(l, tt, r, bb); // B128
  out_ctr[o] = c;
}

extern "C" void kernel_launch(void* const* d_in, const int* in_sizes, int n_in,
                              void* d_out, int out_size, void* d_ws, size_t ws_size,
                              hipStream_t stream) {
  (void)n_in; (void)out_size; (void)d_ws; (void)ws_size;

  const float* locations   = (const float*)d_in[0];
  const float* stride_loc  = (const float*)d_in[1];
  const float* size_ranges = (const float*)d_in[2];
  const float* gt_boxes    = (const float*)d_in[3];
  const int*   gt_classes  = (const int*)d_in[4];

  const int L = in_sizes[1];          // stride_loc has L elements
  const int B = in_sizes[4] / NBOX;   // gt_classes is (B, 50)

  // Outputs concatenated flat in return order: labels (B,L) int32,
  // reg_t (B,L,4) f32, ctr (B,L) f32.
  int*   out_labels = (int*)d_out;
  float* out_reg    = (float*)d_out + (size_t)B * (size_t)L;
  float* out_ctr    = out_reg + (size_t)4 * (size_t)B * (size_t)L;

  dim3 grid((L + BLOCK - 1) / BLOCK, B);
  hipLaunchKernelGGL(fcos_assign_kernel, grid, dim3(BLOCK), 0, stream,
                     locations, stride_loc, size_ranges, gt_boxes, gt_classes,
                     out_labels, out_reg, out_ctr, L);
}